// ScaleTril_57827439673357
// MI455X (gfx1250) — compile-verified
//
#include <hip/hip_runtime.h>
#include <hip/hip_bf16.h>
#include <math.h>

typedef __attribute__((ext_vector_type(2))) float v2f;
typedef __attribute__((ext_vector_type(8))) float v8f;

#define BM 128   // workgroup tile rows
#define BN 64    // workgroup tile cols
#define KB 32    // K panel staged in LDS
#define LSA 36   // LDS row stride (floats): float4-aligned, bank-conflict-free
#define NB 64    // Cholesky block size

static __device__ __forceinline__ v8f wmma4(v2f a, v2f b, v8f c) {
    return __builtin_amdgcn_wmma_f32_16x16x4_f32(
        false, a, false, b, (short)0, c, false, false);
}

// ---------------------------------------------------------------------------
// C(row0.., row0..)  (+)=  P * P^T  over K cols, where P = A(row0.., col0..).
// Workgroup tile: BM x BN. Grid covers M x M region; uniform early-exit keeps
// only blocks touching the lower triangle (by*BN <= bx*BM + BM-1).
// 8 waves; each wave owns a 2x2 quad of 16x16 fp32-WMMA accumulators (32x32),
// so 4 DS fragment loads feed 4 v_wmma_f32_16x16x4_f32 per K=4 step.
// subtract=0: store, 1: C -= result. Row clamps/guards keep EXEC all-ones
// around every WMMA.
// ---------------------------------------------------------------------------
__global__ void syrk_wmma_kernel(const float* __restrict__ A, float* C,
                                 int N, int row0, int col0, int K, int M,
                                 int subtract) {
    int bx = (int)blockIdx.x, by = (int)blockIdx.y;
    if (by > 2 * bx + 1) return;               // uniform: lower blocks only

    __shared__ float As[BM][LSA];
    __shared__ float Bs[BN][LSA];

    int tid  = threadIdx.x;
    int lane = tid & 31;
    int w    = tid >> 5;
    int qr   = w >> 1;            // 0..3 : 32-row quad within 128-row tile
    int qc   = w & 1;             // 0..1 : 32-col quad within 64-col tile
    int half = lane >> 4;         // A/B fragment K-half (ISA 7.12.2 layout)
    int l    = lane & 15;

    v8f acc00 = {}, acc01 = {}, acc10 = {}, acc11 = {};

    const float* Abase = A + (size_t)row0 * N + col0;

    for (int k0 = 0; k0 < K; k0 += KB) {
        __syncthreads();
        // stage A panel: BM x KB as float4 (1024 vec4 -> 4 per thread)
        for (int i = tid; i < BM * (KB / 4); i += 256) {
            int r = i >> 3, c4 = i & 7;
            int rr = bx * BM + r;
            if (rr >= M) rr = M - 1;           // clamp (guarded at store)
            float4 v = *(const float4*)(Abase + (size_t)rr * N + k0 + c4 * 4);
            *(float4*)&As[r][c4 * 4] = v;
        }
        // stage B panel: BN x KB as float4 (512 vec4 -> 2 per thread)
        for (int i = tid; i < BN * (KB / 4); i += 256) {
            int r = i >> 3, c4 = i & 7;
            int rr = by * BN + r;              // always < M (M % 64 == 0)
            float4 v = *(const float4*)(Abase + (size_t)rr * N + k0 + c4 * 4);
            *(float4*)&Bs[r][c4 * 4] = v;
        }
        __syncthreads();

#pragma unroll
        for (int kk = 0; kk < KB; kk += 4) {
            int ka = kk + 2 * half;  // lanes 0-15: K=kk,kk+1 ; 16-31: kk+2,kk+3
            v2f a0, a1, b0, b1;
            a0.x = As[qr * 32 + l][ka];          a0.y = As[qr * 32 + l][ka + 1];
            a1.x = As[qr * 32 + 16 + l][ka];     a1.y = As[qr * 32 + 16 + l][ka + 1];
            b0.x = Bs[qc * 32 + l][ka];          b0.y = Bs[qc * 32 + l][ka + 1];
            b1.x = Bs[qc * 32 + 16 + l][ka];     b1.y = Bs[qc * 32 + 16 + l][ka + 1];
            acc00 = wmma4(a0, b0, acc00);
            acc01 = wmma4(a0, b1, acc01);
            acc10 = wmma4(a1, b0, acc10);
            acc11 = wmma4(a1, b1, acc11);
        }
    }

    // D layout: VGPR v -> M = v (+8 for lanes 16-31), N = lane&15
    int rb = bx * BM + qr * 32 + (half ? 8 : 0);
    int cb = by * BN + qc * 32 + l;
#pragma unroll
    for (int v = 0; v < 8; v++) {
        int r0 = rb + v, r1 = rb + 16 + v;
        if (r0 < M) {
            size_t i0 = (size_t)(row0 + r0) * N + row0 + cb;
            if (subtract) { C[i0] -= acc00[v]; C[i0 + 16] -= acc01[v]; }
            else          { C[i0]  = acc00[v]; C[i0 + 16]  = acc01[v]; }
        }
        if (r1 < M) {
            size_t i1 = (size_t)(row0 + r1) * N + row0 + cb;
            if (subtract) { C[i1] -= acc10[v]; C[i1 + 16] -= acc11[v]; }
            else          { C[i1]  = acc10[v]; C[i1 + 16]  = acc11[v]; }
        }
    }
}

// out[i][i] += softplus(n[i]) (numerically stable)
__global__ void add_diag_kernel(float* out, const float* __restrict__ n, int N) {
    int i = blockIdx.x * blockDim.x + threadIdx.x;
    if (i >= N) return;
    float v  = n[i];
    float sp = (v > 0.0f) ? v + log1pf(expf(-v)) : log1pf(expf(v));
    out[(size_t)i * N + i] += sp;
}

// Unblocked Cholesky of the NB x NB diagonal block at (j0, j0). One workgroup.
__global__ void potrf_diag_kernel(float* C, int N, int j0) {
    __shared__ float Ab[NB][NB + 1];
    int tid = threadIdx.x;
    for (int i = tid; i < NB * NB; i += 256) {
        int r = i >> 6, c = i & 63;
        Ab[r][c] = C[(size_t)(j0 + r) * N + j0 + c];
    }
    __syncthreads();
    for (int j = 0; j < NB; j++) {
        if (tid == 0) Ab[j][j] = sqrtf(Ab[j][j]);
        __syncthreads();
        float d = Ab[j][j];
        for (int i = j + 1 + tid; i < NB; i += 256) Ab[i][j] /= d;
        __syncthreads();
        int rem = NB - 1 - j;
        for (int idx = tid; idx < rem * rem; idx += 256) {
            int r = j + 1 + idx / rem;
            int c = j + 1 + idx % rem;
            if (c <= r) Ab[r][c] -= Ab[r][j] * Ab[c][j];
        }
        __syncthreads();
    }
    for (int i = tid; i < NB * NB; i += 256) {
        int r = i >> 6, c = i & 63;
        C[(size_t)(j0 + r) * N + j0 + c] = Ab[r][c];
    }
}

// Panel solve: rows [row0, row0+nrows) of columns [j0, j0+NB):
// X * L11^T = A21  (forward substitution per row; L11 cached in LDS)
__global__ void trsm_kernel(float* C, int N, int j0, int row0, int nrows) {
    __shared__ float L[NB][NB + 1];
    int tid = threadIdx.x;
    for (int i = tid; i < NB * NB; i += 256) {
        int r = i >> 6, c = i & 63;
        L[r][c] = C[(size_t)(j0 + r) * N + j0 + c];
    }
    __syncthreads();
    int r = row0 + blockIdx.x * 256 + tid;
    if (r >= row0 + nrows) return;
    float* crow = C + (size_t)r * N + j0;
    float x[NB];
#pragma unroll
    for (int j = 0; j < NB; j++) {
        float s = crow[j];
#pragma unroll
        for (int t = 0; t < j; t++) s -= x[t] * L[j][t];
        x[j] = s / L[j][j];
    }
#pragma unroll
    for (int j = 0; j < NB; j++) crow[j] = x[j];
}

// out[i][j] = 0 for j > i (also clears never-fully-initialized upper blocks)
__global__ void zero_upper_kernel(float* out, int N) {
    size_t idx = (size_t)blockIdx.x * blockDim.x + threadIdx.x;
    int row = (int)(idx >> 12);
    int col = (int)(idx & 4095);
    if (col > row) out[idx] = 0.0f;
}

extern "C" void kernel_launch(void* const* d_in, const int* in_sizes, int n_in,
                              void* d_out, int out_size, void* d_ws, size_t ws_size,
                              hipStream_t stream) {
    const float* matrix = (const float*)d_in[0];  // (4096,4096) f32
    const float* nvec   = (const float*)d_in[1];  // (4096,)     f32
    float* out = (float*)d_out;                   // (4096,4096) f32
    const int N = 4096;

    // 1) S = A A^T into lower-triangular blocks of out (fp32 WMMA SYRK)
    dim3 g1(N / BM, N / BN);
    syrk_wmma_kernel<<<g1, 256, 0, stream>>>(matrix, out, N, 0, 0, N, N, 0);

    // 2) S += diag(softplus(n))
    add_diag_kernel<<<(N + 255) / 256, 256, 0, stream>>>(out, nvec, N);

    // 3) blocked right-looking Cholesky, NB = 64
    for (int k = 0; k < N / NB; k++) {
        int j0 = k * NB;
        potrf_diag_kernel<<<1, 256, 0, stream>>>(out, N, j0);
        int row0  = j0 + NB;
        int nrows = N - row0;
        if (nrows > 0) {
            trsm_kernel<<<(nrows + 255) / 256, 256, 0, stream>>>(out, N, j0, row0, nrows);
            dim3 g2((nrows + BM - 1) / BM, nrows / BN);
            // trailing update: C -= L21 * L21^T (same WMMA kernel, K = NB)
            syrk_wmma_kernel<<<g2, 256, 0, stream>>>(out, out, N, row0, j0, NB,
                                                     nrows, 1);
        }
    }

    // 4) zero strictly-upper triangle
    zero_upper_kernel<<<(int)(((size_t)N * N) / 256), 256, 0, stream>>>(out, N);
}